// PositionalEmbeddingLayer_90099823935530
// MI455X (gfx1250) — compile-verified
//
#include <hip/hip_runtime.h>
#include <math.h>
#include <stdint.h>

// ---------------------------------------------------------------------------
// PositionalEmbeddingLayer on MI455X (gfx1250)
// Pure streaming kernel (~270 MB HBM traffic, ~11.6 us floor @ 23.3 TB/s).
// emb[l,d] is batch-invariant -> computed once per thread in registers.
// Tokens are staged through LDS with the CDNA5 async-tensor path
// (global_load_async_to_lds_b128 + s_wait_asynccnt) double-buffered over b.
// ---------------------------------------------------------------------------

namespace {
constexpr int Bsz     = 16;
constexpr int Lsz     = 4096;
constexpr int Dsz     = 512;
constexpr int D4      = Dsz / 4;                 // 128 float4 per row
constexpr int TILE    = 8;                       // rows per block tile
constexpr int THREADS = 256;                     // 8 wave32
constexpr int SLOTS   = TILE * D4 / THREADS;     // 4 float4 slots / thread
constexpr int BPB     = 4;                       // batches per block
}

__global__ __launch_bounds__(THREADS) void pe_embed_kernel(
    const float* __restrict__ tokens,
    const int*   __restrict__ lengths,
    const float* __restrict__ cls,
    float*       __restrict__ out)
{
  __shared__ float4 stage[2][TILE * D4];         // 2 x 16 KB double buffer

  const int t  = threadIdx.x;
  const int b0 = blockIdx.y * BPB;

  const float4* clsv4 = reinterpret_cast<const float4*>(cls);
  float4*       out4  = reinterpret_cast<float4*>(out);

  // ---- dedicated tile: appended row l == L ----
  if (blockIdx.x == (Lsz / TILE)) {
    for (int i = t; i < BPB * D4; i += THREADS) {
      int b = b0 + (i >> 7);
      int c = i & (D4 - 1);
      int lenb = __builtin_amdgcn_readfirstlane(lengths[b]);
      float4 v = make_float4(0.f, 0.f, 0.f, 0.f);
      if (lenb == Lsz) v = clsv4[c];
      out4[(size_t)b * (Lsz + 1) * D4 + (size_t)Lsz * D4 + c] = v;
    }
    return;
  }

  const int l0 = blockIdx.x * TILE;

  // ---- per-thread slot geometry + batch-invariant emb in registers ----
  int    row[SLOTS], col[SLOTS];
  float4 emb[SLOTS], clsr[SLOTS];
#pragma unroll
  for (int s = 0; s < SLOTS; ++s) {
    int j  = t + s * THREADS;                    // float4 index within tile
    row[s] = j >> 7;                             // row in tile (D4 == 128)
    col[s] = j & (D4 - 1);
    int   d  = col[s] * 4;
    float fl = (float)(l0 + row[s]);
    // freq[d] = 100^(-(d - d%2)/D); phase adds pi/2 on odd d.
    // reference: emb = sin(l * (freq + phase))
    float f0 = __powf(100.0f, -(float)(d)     / (float)Dsz);
    float f2 = __powf(100.0f, -(float)(d + 2) / (float)Dsz);
    const float HPI = 1.57079632679489662f;
    emb[s]  = make_float4(sinf(fl * f0), sinf(fl * (f0 + HPI)),
                          sinf(fl * f2), sinf(fl * (f2 + HPI)));
    clsr[s] = clsv4[col[s]];
  }

  // ---- async stage: each thread loads its own SLOTS float4s into LDS ----
  auto issue = [&](int b, int buf) {
#pragma unroll
    for (int s = 0; s < SLOTS; ++s) {
      int j = t + s * THREADS;
      uint32_t lds = (uint32_t)(uintptr_t)&stage[buf][j];
      uint64_t ga  = (uint64_t)(uintptr_t)(tokens +
                       ((size_t)b * Lsz + (size_t)l0) * Dsz) + (uint64_t)j * 16u;
      asm volatile("global_load_async_to_lds_b128 %0, %1, off"
                   :: "v"(lds), "v"(ga) : "memory");
    }
  };

  // ---- consume + store (valid: tok+emb, len row: cls, else: zeros) ----
  auto consume = [&](int b, int buf, int lenb) {
    float4* obase = out4 + (size_t)b * (Lsz + 1) * D4;
#pragma unroll
    for (int s = 0; s < SLOTS; ++s) {
      int l = l0 + row[s];
      float4 v;
      if (l < lenb) {
        float4 tk = stage[buf][t + s * THREADS];
        v = make_float4(tk.x + emb[s].x, tk.y + emb[s].y,
                        tk.z + emb[s].z, tk.w + emb[s].w);
      } else if (l == lenb) {
        v = clsr[s];
      } else {
        v = make_float4(0.f, 0.f, 0.f, 0.f);
      }
      obase[(size_t)l * D4 + col[s]] = v;
    }
  };

  // ---- software-pipelined batch loop, deterministic ASYNCcnt bookkeeping ----
  int  lenCur    = __builtin_amdgcn_readfirstlane(lengths[b0]);
  bool issuedCur = (l0 < lenCur);
  if (issuedCur) issue(b0, 0);

  for (int k = 0; k < BPB; ++k) {
    int  b          = b0 + k;
    int  lenNext    = 0;
    bool issuedNext = false;
    if (k + 1 < BPB) {
      lenNext    = __builtin_amdgcn_readfirstlane(lengths[b + 1]);
      issuedNext = (l0 < lenNext);
      if (issuedNext) issue(b + 1, (k + 1) & 1);
    }
    if (issuedCur) {
      if (issuedNext) {
        // next stage's SLOTS loads may still be in flight; current stage done
        asm volatile("s_wait_asynccnt %0" :: "n"(SLOTS) : "memory");
      } else {
        asm volatile("s_wait_asynccnt 0" ::: "memory");
      }
    }
    consume(b, k & 1, lenCur);
    lenCur    = lenNext;
    issuedCur = issuedNext;
  }
}

extern "C" void kernel_launch(void* const* d_in, const int* in_sizes, int n_in,
                              void* d_out, int out_size, void* d_ws, size_t ws_size,
                              hipStream_t stream) {
  (void)in_sizes; (void)n_in; (void)out_size; (void)d_ws; (void)ws_size;
  const float* tokens  = (const float*)d_in[0];
  const int*   lengths = (const int*)d_in[1];
  const float* cls     = (const float*)d_in[2];
  float*       out     = (float*)d_out;

  dim3 grid(Lsz / TILE + 1, Bsz / BPB);          // 513 x 4 blocks
  pe_embed_kernel<<<grid, THREADS, 0, stream>>>(tokens, lengths, cls, out);
}